// RecurrentLinearAttention_73976516706542
// MI455X (gfx1250) — compile-verified
//
#include <hip/hip_runtime.h>

typedef float v2f __attribute__((ext_vector_type(2)));
typedef float v8f __attribute__((ext_vector_type(8)));

#define N_ 128
#define H_ 32
#define D_ 128
#define M_ 128
#define EPS_ 1e-6f

__device__ __forceinline__ float elu1(float x) {
    // elu(x) + 1 : x>0 -> x+1 ; x<=0 -> exp(x)
    return x > 0.0f ? x + 1.0f : __expf(x);
}

__global__ __launch_bounds__(256)
void rla_step_kernel(const float* __restrict__ query,
                     const float* __restrict__ key_,
                     const float* __restrict__ value,
                     const float* __restrict__ Si,
                     const float* __restrict__ Zi,
                     float* __restrict__ outV,
                     float* __restrict__ outSi,
                     float* __restrict__ outZi) {
    __shared__ __align__(16) float sQ[D_];
    __shared__ __align__(16) float sK[D_];
    __shared__ float sZ;

    const int nh   = blockIdx.x;          // 0..4095  (n*H + h)
    const int lane = threadIdx.x & 31;
    const int wave = threadIdx.x >> 5;    // 0..7 -> M tile

    const int    vecBase = nh * D_;                       // q,k,value,Zi bases
    const size_t siBase  = (size_t)nh * (size_t)(D_ * M_);

    // ---- wave 0: feature maps, Zi_new, normalizer ----
    if (wave == 0) {
        float denom = 0.0f;
        #pragma unroll
        for (int i = lane; i < D_; i += 32) {
            float qf = elu1(query[vecBase + i]);
            float kf = elu1(key_[vecBase + i]);
            float zn = Zi[vecBase + i] + kf;
            sQ[i] = qf;
            sK[i] = kf;
            outZi[vecBase + i] = zn;
            denom += qf * zn;
        }
        #pragma unroll
        for (int off = 16; off > 0; off >>= 1)
            denom += __shfl_xor(denom, off, 32);
        if (lane == 0) sZ = 1.0f / (denom + EPS_);
    }
    __syncthreads();

    const int hi   = lane >> 4;   // 0: lanes 0-15, 1: lanes 16-31
    const int c    = lane & 15;   // column within 16-wide tile
    const int mcol = wave * 16;   // this wave's M tile base

    // B operand (4x16 f32): row K=0 holds value[mcol..mcol+15]; rows 1..3 zero.
    // Layout: VGPR0 lanes 0-15 -> K=0, lanes 16-31 -> K=2; VGPR1 -> K=1/K=3.
    v2f B;
    B.x = (hi == 0) ? value[vecBase + mcol + c] : 0.0f;
    B.y = 0.0f;

    float vacc = 0.0f;

    const float* __restrict__ siSrc = Si    + siBase;
    float*       __restrict__ siDst = outSi + siBase;

    #pragma unroll 1
    for (int d = 0; d < D_ / 16; ++d) {
        if (d + 1 < D_ / 16)
            __builtin_prefetch(siSrc + (size_t)((d + 1) * 16) * M_ + mcol + c, 0, 1);

        // A operand (16x4 f32): column K=0 holds Kf[d*16 + row]; K=1..3 zero.
        // Layout: VGPR0 lanes 0-15 -> (M=lane, K=0), lanes 16-31 -> K=2.
        v2f A;
        A.x = (hi == 0) ? sK[d * 16 + lane] : 0.0f;
        A.y = 0.0f;

        // C operand: 16x16 Si tile, rows [d*16, d*16+16), cols [mcol, mcol+16)
        // C layout: VGPR j, lane -> element (row = j + 8*hi, col = c)
        const size_t tbase = (size_t)(d * 16 + 8 * hi) * M_ + (size_t)(mcol + c);
        v8f C;
        #pragma unroll
        for (int j = 0; j < 8; ++j)
            C[j] = siSrc[tbase + (size_t)j * M_];

        // Si_new tile = Kf_tile (x) value_tile + Si tile  -- one matrix op
        v8f Dt = __builtin_amdgcn_wmma_f32_16x16x4_f32(
            false, A, false, B, (short)0, C, false, false);

        // stream Si_new out + fold Qf^T * Si_new into V partials
        const float4 q0 = ((const float4*)sQ)[d * 4 + 2 * hi + 0];
        const float4 q1 = ((const float4*)sQ)[d * 4 + 2 * hi + 1];
        #pragma unroll
        for (int j = 0; j < 8; ++j)
            siDst[tbase + (size_t)j * M_] = Dt[j];
        vacc += q0.x * Dt[0] + q0.y * Dt[1] + q0.z * Dt[2] + q0.w * Dt[3];
        vacc += q1.x * Dt[4] + q1.y * Dt[5] + q1.z * Dt[6] + q1.w * Dt[7];
    }

    // lanes l and l^16 hold complementary row-halves for column c
    vacc += __shfl_xor(vacc, 16, 32);
    if (hi == 0)
        outV[nh * M_ + mcol + c] = vacc * sZ;
}

extern "C" void kernel_launch(void* const* d_in, const int* in_sizes, int n_in,
                              void* d_out, int out_size, void* d_ws, size_t ws_size,
                              hipStream_t stream) {
    const float* query = (const float*)d_in[0];
    const float* key_  = (const float*)d_in[1];
    const float* value = (const float*)d_in[2];
    const float* Si    = (const float*)d_in[3];
    const float* Zi    = (const float*)d_in[4];

    float* outV  = (float*)d_out;                          // [N,H,M]
    float* outSi = outV  + (size_t)N_ * H_ * M_;           // [N,H,D,M]
    float* outZi = outSi + (size_t)N_ * H_ * D_ * M_;      // [N,H,D]

    dim3 grid(N_ * H_);   // 4096 workgroups, one per (n,h)
    dim3 block(256);      // 8 wave32 = 8 M-tiles
    hipLaunchKernelGGL(rla_step_kernel, grid, block, 0, stream,
                       query, key_, value, Si, Zi, outV, outSi, outZi);
}